// DeepGCNFeatureNet_36962488549947
// MI455X (gfx1250) — compile-verified
//
#include <hip/hip_runtime.h>
#include <math.h>

typedef __attribute__((ext_vector_type(2))) float v2f;
typedef __attribute__((ext_vector_type(8))) float v8f;

#define T_PTS   100
#define T_PAD   112         // 7 * 16 (gram tiling)
#define K_NBR   8
#define CIN     4
#define COUT    64
#define KPAD    20          // 18 edge channels padded to 20 (5 k-steps of 4)
#define NTILES  4           // 64 / 16
#define MTILES  50          // 800 / 16
#define GTILES  49          // 7 x 7 gram tiles
#define GPITCH  104         // cols 100..103 are write-only padding
#define VXs     0.2f
#define VYs     0.2f
#define XOFF    (0.2f * 0.5f + 0.0f)
#define YOFF    (0.2f * 0.5f + (-40.0f))
#define BN_EPS  0.001f
#define SLOPE   0.2f

// Monotone float <-> signed-int encoding so shared-memory atomicMax(int) is a
// deterministic float max (order-independent reduction).
__device__ __forceinline__ int f2ord(float f) {
    int i = __float_as_int(f);
    return (i >= 0) ? i : (i ^ 0x7fffffff);
}
__device__ __forceinline__ float ord2f(int i) {
    return __int_as_float((i >= 0) ? i : (i ^ 0x7fffffff));
}

__global__ __launch_bounds__(256)
void dgcn_pillar_kernel(const float* __restrict__ features,   // [V,100,4]
                        const float* __restrict__ conv_w,     // [64,18]
                        const float* __restrict__ bn_gamma,   // [64]
                        const float* __restrict__ bn_beta,    // [64]
                        const float* __restrict__ bn_mean,    // [64]
                        const float* __restrict__ bn_var,     // [64]
                        const int*   __restrict__ num_voxels, // [V]
                        const int*   __restrict__ coors,      // [V,4]
                        float*       __restrict__ out)        // [V,64]
{
    __shared__ float feat[T_PAD][12];        // 9 channels (stride 12), rows 100..111 zero
    __shared__ float xxs[T_PTS];
    __shared__ int   idxs[T_PTS][K_NBR];
    __shared__ float wsc[KPAD][COUT];        // BN-scale-folded W^T, k-padded
    __shared__ float betap[COUT];            // beta - mean*scale
    __shared__ int   omax[COUT];             // ordered-int running max over T
    __shared__ float mean3[3];
    __shared__ float gram[T_PTS + 1][GPITCH];// row 100 + cols 100..103 = trash cells

    const int n   = blockIdx.x;
    const int tid = threadIdx.x;
    const int nv  = num_voxels[n];
    const float* fbase = features + (size_t)n * T_PTS * CIN;

    const int lane  = tid & 31;
    const int lrow  = lane & 15;       // A: M index; B/C: N index (mod 16)
    const int khalf = lane >> 4;       // 0 -> K{0,1}, 1 -> K{2,3} within a k-step
    const int wave  = tid >> 5;        // 8 wave32s

    // ---- phase 1: raw feature load + weight prep + accumulator init ----
    for (int i = tid; i < T_PTS * CIN; i += 256) {
        feat[i >> 2][i & 3] = fbase[i];
    }
    // zero padding rows 100..111 (all 12 channels) so gram tiles are clean
    if (tid < (T_PAD - T_PTS) * 12) {
        feat[T_PTS + tid / 12][tid % 12] = 0.0f;
    }
    for (int i = tid; i < KPAD * COUT; i += 256) {
        int kr = i >> 6, o = i & 63;
        float sc = bn_gamma[o] / sqrtf(bn_var[o] + BN_EPS);
        wsc[kr][o] = (kr < 18) ? conv_w[o * 18 + kr] * sc : 0.0f;
    }
    if (tid < COUT) {
        float sc = bn_gamma[tid] / sqrtf(bn_var[tid] + BN_EPS);
        betap[tid] = bn_beta[tid] - bn_mean[tid] * sc;
        omax[tid]  = f2ord(-INFINITY);
    }
    __syncthreads();

    // ---- phase 2: points_mean = sum over ALL T rows (unmasked) / nv ----
    if (tid < 3) {
        float s = 0.0f;
        for (int t = 0; t < T_PTS; ++t) s += feat[t][tid];
        mean3[tid] = s / (float)nv;
    }
    __syncthreads();

    // ---- phase 3: build masked 9-channel features + squared norms ----
    if (tid < T_PTS) {
        const int t = tid;
        const float m  = (t < nv) ? 1.0f : 0.0f;
        const float f0 = feat[t][0], f1 = feat[t][1];
        const float f2 = feat[t][2], f3 = feat[t][3];
        const float cx = (float)coors[n * 4 + 3] * VXs + XOFF;
        const float cy = (float)coors[n * 4 + 2] * VYs + YOFF;
        float r[9];
        r[0] = f0; r[1] = f1; r[2] = f2; r[3] = f3;
        r[4] = f0 - mean3[0]; r[5] = f1 - mean3[1]; r[6] = f2 - mean3[2];
        r[7] = f0 - cx; r[8] = f1 - cy;
        float xx = 0.0f;
        #pragma unroll
        for (int c = 0; c < 9; ++c) {
            float v = r[c] * m;
            feat[t][c] = v;
            xx += v * v;
        }
        feat[t][9] = 0.0f; feat[t][10] = 0.0f; feat[t][11] = 0.0f;
        xxs[t] = xx;
    }
    __syncthreads();

    // ---- phase 4a: Gram matrix  G = X . X^T  via V_WMMA_F32_16X16X4_F32 ----
    // X is feat[112][12]; B = X^T so B[k][n] = feat[n][k] -> both fragments are
    // plain row reads of feat. 7x7 tiles x 3 k-steps, spread over 8 waves.
    for (int tix = wave; tix < GTILES; tix += 8) {
        const int mt = tix / 7, nt = tix % 7;
        const int arow = 16 * mt + lrow;
        const int brow = 16 * nt + lrow;
        v8f g = {0.f, 0.f, 0.f, 0.f, 0.f, 0.f, 0.f, 0.f};
        #pragma unroll
        for (int ks = 0; ks < 3; ++ks) {
            const int cb = 4 * ks + 2 * khalf;
            v2f a, b;
            a.x = feat[arow][cb]; a.y = feat[arow][cb + 1];
            b.x = feat[brow][cb]; b.y = feat[brow][cb + 1];
            g = __builtin_amdgcn_wmma_f32_16x16x4_f32(
                false, a, false, b, (short)0, g, false, false);
        }
        // C layout: lane holds column s = brow, rows t = 16*mt + 8*khalf + r.
        // Out-of-range (padding) results go to trash cells via clamping instead
        // of exec-mask branches: row 100 and cols 100..103 are never read.
        const int s_c = (brow < T_PTS) ? brow : (GPITCH - 1);
        const int t_b = 16 * mt + 8 * khalf;
        #pragma unroll
        for (int r = 0; r < 8; ++r) {
            const int t   = t_b + r;
            const int t_c = (t < T_PTS) ? t : T_PTS;
            gram[t_c][s_c] = g[r];
        }
    }
    __syncthreads();

    // ---- phase 4b: stable top-8 of neg_dist (ties -> smaller index) ----
    if (tid < T_PTS) {
        const int t = tid;
        const float xt = xxs[t];
        float bv[K_NBR]; int bi[K_NBR];
        #pragma unroll
        for (int k = 0; k < K_NBR; ++k) { bv[k] = -INFINITY; bi[k] = 0; }
        for (int s = 0; s < T_PTS; ++s) {
            float nd = 2.0f * gram[t][s] - xt - xxs[s];
            if (nd > bv[K_NBR - 1]) {       // strict >: stable vs earlier index
                int p = K_NBR - 1;
                while (p > 0 && nd > bv[p - 1]) {
                    bv[p] = bv[p - 1]; bi[p] = bi[p - 1]; --p;
                }
                bv[p] = nd; bi[p] = s;
            }
        }
        #pragma unroll
        for (int k = 0; k < K_NBR; ++k) idxs[t][k] = bi[k];
    }
    __syncthreads();

    // ---- phase 5: EdgeConv GEMM via V_WMMA_F32_16X16X4_F32 ----
    // edge rows: row = t*8 + k  (800 rows, 50 M-tiles), cols: 18 -> pad 20.

    // Preload B fragments (W^T with BN scale folded), all 5 k-steps x 4 n-tiles.
    // B layout (4x16 f32): VGPR0 = row K=2*khalf, VGPR1 = row K=2*khalf+1.
    v2f bfrag[5][NTILES];
    #pragma unroll
    for (int ks = 0; ks < 5; ++ks) {
        #pragma unroll
        for (int nt = 0; nt < NTILES; ++nt) {
            int kr = 4 * ks + 2 * khalf;
            int nn = 16 * nt + lrow;
            bfrag[ks][nt].x = wsc[kr][nn];
            bfrag[ks][nt].y = wsc[kr + 1][nn];
        }
    }

    auto edgech = [&](int c, int t, int j) -> float {
        if (c < 9)  return feat[j][c] - feat[t][c];   // nbr - xc
        if (c < 18) return feat[t][c - 9];            // xc
        return 0.0f;                                  // k-padding
    };

    for (int mt = wave; mt < MTILES; mt += 8) {
        // A-fragment row for this lane (A layout: M = lane%16 in both halves)
        const int row = 16 * mt + lrow;
        const int t_a = row >> 3;
        const int j_a = idxs[t_a][row & 7];

        v8f acc[NTILES];
        #pragma unroll
        for (int nt = 0; nt < NTILES; ++nt) {
            float b0 = betap[16 * nt + lrow];         // depends only on N
            #pragma unroll
            for (int r = 0; r < 8; ++r) acc[nt][r] = b0;
        }

        #pragma unroll
        for (int ks = 0; ks < 5; ++ks) {
            const int cb = 4 * ks + 2 * khalf;        // A: lanes 0-15 K0/K1, 16-31 K2/K3
            v2f a;
            a.x = edgech(cb,     t_a, j_a);
            a.y = edgech(cb + 1, t_a, j_a);
            #pragma unroll
            for (int nt = 0; nt < NTILES; ++nt) {
                acc[nt] = __builtin_amdgcn_wmma_f32_16x16x4_f32(
                    false, a, false, bfrag[ks][nt],
                    (short)0, acc[nt], false, false);
            }
        }

        // C layout: this lane's 8 components are rows M = 8*khalf + r, i.e. the
        // 8 k-neighbors of point t = 2*mt + khalf  ->  max-over-K is free.
        const int t_c = 2 * mt + khalf;
        const float msk = (t_c < nv) ? 1.0f : 0.0f;   // out * mask before T-max
        #pragma unroll
        for (int nt = 0; nt < NTILES; ++nt) {
            float mx = -INFINITY;
            #pragma unroll
            for (int r = 0; r < 8; ++r) {
                float v = acc[nt][r];
                v = (v > 0.0f) ? v : SLOPE * v;       // leaky relu
                mx = fmaxf(mx, v);
            }
            atomicMax(&omax[16 * nt + lrow], f2ord(mx * msk));
        }
    }
    __syncthreads();

    // ---- phase 6: write per-voxel 64-channel result ----
    if (tid < COUT) out[(size_t)n * COUT + tid] = ord2f(omax[tid]);
}

extern "C" void kernel_launch(void* const* d_in, const int* in_sizes, int n_in,
                              void* d_out, int out_size, void* d_ws, size_t ws_size,
                              hipStream_t stream) {
    const float* features   = (const float*)d_in[0];
    const float* conv_w     = (const float*)d_in[1];
    const float* bn_gamma   = (const float*)d_in[2];
    const float* bn_beta    = (const float*)d_in[3];
    const float* bn_mean    = (const float*)d_in[4];
    const float* bn_var     = (const float*)d_in[5];
    const int*   num_voxels = (const int*)d_in[6];
    const int*   coors      = (const int*)d_in[7];
    float*       out        = (float*)d_out;

    const int V = in_sizes[6];   // one workgroup per voxel
    dgcn_pillar_kernel<<<V, 256, 0, stream>>>(
        features, conv_w, bn_gamma, bn_beta, bn_mean, bn_var,
        num_voxels, coors, out);
}